// PairwiseEmbedding_31963146617204
// MI455X (gfx1250) — compile-verified
//
#include <hip/hip_runtime.h>
#include <hip/hip_bf16.h>
#include <math.h>

// ---------------------------------------------------------------------------
// PairwiseEmbedding for gfx1250 (MI455X).
// Feature layout per edge (340 f32):
//   [0]        noised_dst
//   [1]        noised_src
//   [2..22]    src one-hot (21) * keep_src
//   [23..43]   dst one-hot (21) * keep_dst
//   [44..299]  RBF(dists 4x4 -> 16 each) = 256
//   [300..315] positional cos(8)+sin(8)
//   [316..327] src_bb_local (12) * keep_src
//   [328..339] dst_in_src   (12) * keep_dst
// ---------------------------------------------------------------------------

#define EPSF    1e-8f
#define F_TOT   340
#define RES_STR 48   // floats per residue in workspace
// ws layout per residue:
//  [0..11]  bb (N,CA,C,CBvirt xyz)
//  [12..23] local coords (R^T (bb - t)), 12
//  [24..32] R row-major R[j*3+i]
//  [33..35] t
//  [36..39] asq[k] = |bb_k + eps|^2   (src-side norms)
//  [40..43] bsq[k] = |bb_k|^2         (dst-side norms)
//  [44..47] pad

// 0.8 * sqrt(log2(e)) : exp(-((d-mu)*0.8)^2) == exp2(-((d-mu)*RBF_S)^2)
#define RBF_S 0.9608979270291598f

typedef float v2f __attribute__((ext_vector_type(2)));
typedef float v8f __attribute__((ext_vector_type(8)));

__device__ __constant__ float c_freq[8] = {
    1.0f, 0.316227766016837933f, 0.1f, 0.0316227766016837933f,
    0.01f, 0.00316227766016837933f, 0.001f, 0.000316227766016837933f};

__device__ __forceinline__ int sel4i(int a, int b, int c, int d, int s) {
  int r = a;
  r = (s == 1) ? b : r;
  r = (s == 2) ? c : r;
  r = (s == 3) ? d : r;
  return r;
}
__device__ __forceinline__ float sel4f(float a, float b, float c, float d, int s) {
  float r = a;
  r = (s == 1) ? b : r;
  r = (s == 2) ? c : r;
  r = (s == 3) ? d : r;
  return r;
}

// ---------------------------------------------------------------------------
// Kernel A: per-residue precompute (tables land in L2; 3.8 MB total)
// ---------------------------------------------------------------------------
__global__ __launch_bounds__(256) void pe_residue_prep(
    const float* __restrict__ atom14, const float* __restrict__ rigids7,
    float* __restrict__ ws, int n_res) {
  int r = blockIdx.x * 256 + threadIdx.x;
  if (r >= n_res) return;

  const float* A = atom14 + (long)r * 42;  // 14 atoms x 3
  float n0 = A[0], n1 = A[1], n2 = A[2];
  float a0 = A[3], a1 = A[4], a2 = A[5];
  float c0 = A[6], c1 = A[7], c2 = A[8];
  float b0 = a0 - n0, b1 = a1 - n1, b2 = a2 - n2;
  float e0 = c0 - a0, e1 = c1 - a1, e2 = c2 - a2;
  float x0 = b1 * e2 - b2 * e1;
  float x1 = b2 * e0 - b0 * e2;
  float x2 = b0 * e1 - b1 * e0;
  float cb0 = -0.58273431f * x0 + 0.56802827f * b0 - 0.54067466f * e0 + a0;
  float cb1 = -0.58273431f * x1 + 0.56802827f * b1 - 0.54067466f * e1 + a1;
  float cb2 = -0.58273431f * x2 + 0.56802827f * b2 - 0.54067466f * e2 + a2;

  float bb[12] = {n0, n1, n2, a0, a1, a2, c0, c1, c2, cb0, cb1, cb2};

  const float* q = rigids7 + (long)r * 7;
  float qw = q[0], qx = q[1], qy = q[2], qz = q[3];
  float inv = 1.0f / sqrtf(qw * qw + qx * qx + qy * qy + qz * qz);
  qw *= inv; qx *= inv; qy *= inv; qz *= inv;
  float R[9];
  R[0] = 1.f - 2.f * (qy * qy + qz * qz);
  R[1] = 2.f * (qx * qy - qw * qz);
  R[2] = 2.f * (qx * qz + qw * qy);
  R[3] = 2.f * (qx * qy + qw * qz);
  R[4] = 1.f - 2.f * (qx * qx + qz * qz);
  R[5] = 2.f * (qy * qz - qw * qx);
  R[6] = 2.f * (qx * qz - qw * qy);
  R[7] = 2.f * (qy * qz + qw * qx);
  R[8] = 1.f - 2.f * (qx * qx + qy * qy);
  float t0 = q[4], t1 = q[5], t2 = q[6];

  float* W = ws + (long)r * RES_STR;
#pragma unroll
  for (int i = 0; i < 12; i++) W[i] = bb[i];
#pragma unroll
  for (int k = 0; k < 4; k++) {
    float p0 = bb[k * 3] - t0, p1 = bb[k * 3 + 1] - t1, p2 = bb[k * 3 + 2] - t2;
    // local[k][i] = sum_j R[j][i] * p_j
    W[12 + k * 3 + 0] = R[0] * p0 + R[3] * p1 + R[6] * p2;
    W[12 + k * 3 + 1] = R[1] * p0 + R[4] * p1 + R[7] * p2;
    W[12 + k * 3 + 2] = R[2] * p0 + R[5] * p1 + R[8] * p2;
  }
#pragma unroll
  for (int i = 0; i < 9; i++) W[24 + i] = R[i];
  W[33] = t0; W[34] = t1; W[35] = t2;
#pragma unroll
  for (int k = 0; k < 4; k++) {
    float s0 = bb[k * 3] + EPSF, s1 = bb[k * 3 + 1] + EPSF, s2 = bb[k * 3 + 2] + EPSF;
    W[36 + k] = s0 * s0 + s1 * s1 + s2 * s2;
    float u0 = bb[k * 3], u1 = bb[k * 3 + 1], u2 = bb[k * 3 + 2];
    W[40 + k] = u0 * u0 + u1 * u1 + u2 * u2;
  }
}

// ---------------------------------------------------------------------------
// Kernel B: one wave32 per 4 edges. WMMA f32 16x16x4 computes the
// (-2 * src_pt . dst_pt) dot blocks for 4 edges in one matrix op.
//   A rows m=4g+i : -2*(bb_src[g][i]+eps)    (16x4, K=coords, K3=0)
//   B cols n=4g+j :      bb_dst[g][j]        (4x16)
//   D[m][n] useful only on 4x4 block diagonal.
// dist^2(g,i,j) = asq[g][i] + bsq[g][j] + D[4g+i][4g+j]
// ---------------------------------------------------------------------------
__global__ __launch_bounds__(256) void pe_edge_feats(
    const float* __restrict__ ws, const int* __restrict__ seq,
    const unsigned char* __restrict__ nmask, const int* __restrict__ eidx,
    float* __restrict__ out, int n_edge) {
  const int lane = threadIdx.x & 31;
  int wid = blockIdx.x * (blockDim.x >> 5) + (threadIdx.x >> 5);
  wid = __builtin_amdgcn_readfirstlane(wid);
  const int ebase = wid * 4;
  if (ebase >= n_edge) return;          // wave-uniform
  const int ne = (n_edge - ebase) < 4 ? (n_edge - ebase) : 4;

  // --- per-edge scalars (wave-uniform) ---
  int dstv[4], srcv[4], sq_s[4], sq_d[4];
  float nm_s[4], nm_d[4], kp_s[4], kp_d[4], dfl[4];
#pragma unroll
  for (int e = 0; e < 4; e++) {
    int ei = ebase + e;
    ei = ei < n_edge ? ei : n_edge - 1;
    dstv[e] = eidx[ei];
    srcv[e] = eidx[n_edge + ei];
    sq_s[e] = seq[srcv[e]];
    sq_d[e] = seq[dstv[e]];
    nm_s[e] = nmask[srcv[e]] ? 1.f : 0.f;
    nm_d[e] = nmask[dstv[e]] ? 1.f : 0.f;
    kp_s[e] = 1.f - nm_s[e];
    kp_d[e] = 1.f - nm_d[e];
    dfl[e] = (float)(dstv[e] - srcv[e]);
  }

  // --- pack A (src, scaled by -2, +eps) and B (dst) per ISA 16x4 f32 layout ---
  const bool hi = lane >= 16;
  const int idx = lane & 15;
  const int eg = idx >> 2;  // edge-in-group for this row/col
  const int pt = idx & 3;   // backbone point index

  const int rs = sel4i(srcv[0], srcv[1], srcv[2], srcv[3], eg);
  const int rd = sel4i(dstv[0], dstv[1], dstv[2], dstv[3], eg);
  const float* Ps = ws + (long)rs * RES_STR + pt * 3;
  const float* Pd = ws + (long)rd * RES_STR + pt * 3;
  float sx = Ps[0], sy = Ps[1], sz = Ps[2];
  float dx = Pd[0], dy = Pd[1], dz = Pd[2];

  v2f av, bv;
  av.x = hi ? -2.f * (sz + EPSF) : -2.f * (sx + EPSF);  // K=2 | K=0
  av.y = hi ? 0.f : -2.f * (sy + EPSF);                 // K=3 | K=1
  bv.x = hi ? dz : dx;
  bv.y = hi ? 0.f : dy;

  v8f cz = {};
  v8f dmat = __builtin_amdgcn_wmma_f32_16x16x4_f32(
      false, av, false, bv, (short)0, cz, false, false);

  // --- distances: lane (e_loc,j) holds dist(e_loc, i=0..3, j) ---
  // useful lanes: (!hi && idx<8) -> groups 0,1 ; (hi && idx>=8) -> groups 2,3
  const float* As = ws + (long)rs * RES_STR + 36;
  float asq0 = As[0], asq1 = As[1], asq2 = As[2], asq3 = As[3];
  float bsq = ws[(long)rd * RES_STR + 40 + pt];
  const bool bhigh = (eg & 1) != 0;  // rows 4..7 (or 12..15) -> dmat[4..7]
  float q0 = bhigh ? dmat[4] : dmat[0];
  float q1 = bhigh ? dmat[5] : dmat[1];
  float q2 = bhigh ? dmat[6] : dmat[2];
  float q3 = bhigh ? dmat[7] : dmat[3];
  // raw v_sqrt_f32 (~1 ulp): ample for a distance feeding a Gaussian
  float dist0 = __builtin_amdgcn_sqrtf(fmaxf(asq0 + bsq + q0, 0.f));
  float dist1 = __builtin_amdgcn_sqrtf(fmaxf(asq1 + bsq + q1, 0.f));
  float dist2 = __builtin_amdgcn_sqrtf(fmaxf(asq2 + bsq + q2, 0.f));
  float dist3 = __builtin_amdgcn_sqrtf(fmaxf(asq3 + bsq + q3, 0.f));

  // --- RBF: 256 coalesced floats per edge, 8 store rounds of 32 lanes ---
  // value = exp2(-((dv - mu)*RBF_S)^2) ; mu = linspace(2,22,16)[lane&15]
  const float mus = (2.0f + (float)(lane & 15) * (4.0f / 3.0f)) * RBF_S;
  const int h = hi ? 1 : 0;
#pragma unroll
  for (int e = 0; e < 4; e++) {
    if (e >= ne) break;  // wave-uniform
    float* op = out + (size_t)(ebase + e) * F_TOT + 44;
#pragma unroll
    for (int it = 0; it < 8; it++) {
      const int dcode = 2 * it + h;      // dist index 0..15 = i*4+j
      const int jj = dcode & 3;
      const int sl = (e < 2) ? (4 * e + jj) : (16 + 4 * e + jj);
      float dsrc;
      switch (it >> 1) {                 // compile-time after unroll
        case 0: dsrc = dist0; break;
        case 1: dsrc = dist1; break;
        case 2: dsrc = dist2; break;
        default: dsrc = dist3; break;
      }
      float dv = __shfl(dsrc, sl, 32);
      float tt = __builtin_fmaf(dv, RBF_S, -mus);
      op[it * 32 + lane] = __builtin_amdgcn_exp2f(-(tt * tt));
    }
  }

  // --- flags + one-hots: features [0..43] per edge ---
#pragma unroll
  for (int e = 0; e < 4; e++) {
    if (e >= ne) break;  // wave-uniform
    float* op = out + (size_t)(ebase + e) * F_TOT;
#pragma unroll
    for (int r0 = 0; r0 < 2; r0++) {
      int uu = r0 * 32 + lane;
      if (uu < 44) {
        float val;
        if (uu == 0) val = nm_d[e];
        else if (uu == 1) val = nm_s[e];
        else if (uu < 23) val = (sq_s[e] == uu - 2) ? kp_s[e] : 0.f;
        else val = (sq_d[e] == uu - 23) ? kp_d[e] : 0.f;
        op[uu] = val;
      }
    }
  }

  // --- positional: one full-wave precise sincos round (4 edges x 8 freqs) ---
  {
    const int e = lane >> 3;  // 0..3
    const int p = lane & 7;   // freq index
    float dflv = sel4f(dfl[0], dfl[1], dfl[2], dfl[3], e);
    float ang = dflv * c_freq[p];
    float s, c;
    sincosf(ang, &s, &c);  // precise: |ang| up to ~2e4 rad
    if (e < ne) {
      float* op = out + (size_t)(ebase + e) * F_TOT + 300;
      op[p] = c;
      op[8 + p] = s;
    }
  }

  // --- local frames: 2 full-wave rounds, 16 lanes per edge, 12 used ---
#pragma unroll
  for (int r2 = 0; r2 < 2; r2++) {
    const int e = 2 * r2 + (lane >> 4);  // 0..3
    const int fr = lane & 15;
    const int rsL = sel4i(srcv[0], srcv[1], srcv[2], srcv[3], e);
    const int rdL = sel4i(dstv[0], dstv[1], dstv[2], dstv[3], e);
    const float kps = sel4f(kp_s[0], kp_s[1], kp_s[2], kp_s[3], e);
    const float kpd = sel4f(kp_d[0], kp_d[1], kp_d[2], kp_d[3], e);
    const float* WR = ws + (long)rsL * RES_STR;
    const float* WD = ws + (long)rdL * RES_STR;
    if (fr < 12 && e < ne) {
      float* op = out + (size_t)(ebase + e) * F_TOT;
      // src_bb_local * keep_src
      op[316 + fr] = WR[12 + fr] * kps;
      // dst_in_src[k][i] = sum_j Rsrc[j][i]*(bb_dst[k][j]-t_src[j]), * keep_dst
      int kk = (fr >= 9) ? 3 : (fr >= 6) ? 2 : (fr >= 3) ? 1 : 0;
      int ii = fr - 3 * kk;
      float v = 0.f;
#pragma unroll
      for (int j = 0; j < 3; j++)
        v += WR[24 + j * 3 + ii] * (WD[kk * 3 + j] - WR[33 + j]);
      op[328 + fr] = v * kpd;
    }
  }
}

// ---------------------------------------------------------------------------
extern "C" void kernel_launch(void* const* d_in, const int* in_sizes, int n_in,
                              void* d_out, int out_size, void* d_ws, size_t ws_size,
                              hipStream_t stream) {
  const float* atom14 = (const float*)d_in[0];
  const float* rigids7 = (const float*)d_in[1];
  const int* seq = (const int*)d_in[2];
  const unsigned char* nmask = (const unsigned char*)d_in[3];  // jnp.bool_
  const int* eidx = (const int*)d_in[4];
  float* out = (float*)d_out;
  float* ws = (float*)d_ws;

  const int n_res = in_sizes[2];
  const int n_edge = in_sizes[4] / 2;

  int blkA = (n_res + 255) / 256;
  pe_residue_prep<<<blkA, 256, 0, stream>>>(atom14, rigids7, ws, n_res);

  int nwaves = (n_edge + 3) / 4;
  int blkB = (nwaves + 7) / 8;  // 8 waves (256 threads) per block
  pe_edge_feats<<<blkB, 256, 0, stream>>>(ws, seq, nmask, eidx, out, n_edge);
}